// DifferentiateAttention_70875550319397
// MI455X (gfx1250) — compile-verified
//
#include <hip/hip_runtime.h>

typedef __attribute__((ext_vector_type(2))) float v2f;
typedef __attribute__((ext_vector_type(8))) float v8f;

constexpr int Bq = 128, Rr = 29, Dd = 1024, Ah = 8, Nn = 1024;
constexpr int NT    = 16;          // keys per tile
constexpr int WAVES = 8;           // waves per block (wave32)
constexpr int DPW   = Dd / WAVES;  // 128 d-columns owned per wave
constexpr int YSTR  = Dd + (Dd >> 8);  // 1028: +1 DWORD pad per 256 DWORDs
constexpr int TILEF = NT * YSTR;       // floats per Y buffer (16448)
constexpr int NTILES = Nn / NT;        // 64

#if defined(__has_builtin)
#if __has_builtin(__builtin_amdgcn_tensor_load_to_lds)
#define HAVE_TDM 1
#endif
#endif
#ifndef HAVE_TDM
#define HAVE_TDM 0
#endif

// scl[a,d] = wx[a,d,d]*wy[a,d,d]*wx_bias[d]*wy_bias[d] / sqrt(D)
__global__ void diffattn_scale_kernel(const float* __restrict__ wx,
                                      const float* __restrict__ wxb,
                                      const float* __restrict__ wy,
                                      const float* __restrict__ wyb,
                                      float* __restrict__ scl) {
  int idx = blockIdx.x * blockDim.x + threadIdx.x;
  if (idx >= Ah * Dd) return;
  int a = idx >> 10, d = idx & (Dd - 1);
  size_t diag = (size_t)a * Dd * Dd + (size_t)d * Dd + d;
  scl[idx] = wx[diag] * wy[diag] * wxb[d] * wyb[d] * 0.03125f; // 1/sqrt(1024)
}

#if HAVE_TDM
// Issue one TDM load of a contiguous 16x1024 f32 tile into LDS at
// lds_byte_off, inserting 1 pad DWORD every 256 DWORDs (-> row stride 1028).
// D# bitfields per CDNA5 ISA ch.8.
__device__ __forceinline__ void tdm_load_tile(const float* gsrc,
                                              unsigned lds_byte_off) {
  typedef unsigned int t4u __attribute__((ext_vector_type(4)));
  typedef int t8i __attribute__((ext_vector_type(8)));
  typedef int t4i __attribute__((ext_vector_type(4)));
  unsigned long long ga = (unsigned long long)(size_t)gsrc;
  t4u g0;
  g0.x = 1u;                                  // count=1, user D#, no gather
  g0.y = lds_byte_off;                        // lds_addr
  g0.z = (unsigned)(ga & 0xFFFFFFFFu);        // global_addr[31:0]
  g0.w = (unsigned)((ga >> 32) & 0x1FFFFFFu)  // global_addr[56:32]
         | (2u << 30);                        // type=2 ("image")
  t8i g1;
  g1[0] = (int)((2u << 16)     // data_size = 4 bytes
                | (1u << 20)   // pad_enable
                | (7u << 22)); // pad_interval: every 256 DWORDs (+1 DWORD)
  g1[1] = (int)(1024u << 16);  // tensor_dim0 = 1024  (bits 79:48, low half)
  g1[2] = (int)(1024u << 16);  // tensor_dim1 = 1024  (bits 111:80, low half)
  g1[3] = (int)(1024u << 16);  // tile_dim0   = 1024  (bits 127:112)
  g1[4] = 16;                  // tile_dim1   = 16    (bits 143:128)
  g1[5] = 1024;                // tensor_dim0_stride = 1024 elements
  g1[6] = 0;
  g1[7] = 0;
  t4i z4 = {0, 0, 0, 0};
#if __clang_major__ >= 23
  t8i z8 = {0, 0, 0, 0, 0, 0, 0, 0};
  __builtin_amdgcn_tensor_load_to_lds(g0, g1, z4, z4, z8, 0);
#else
  __builtin_amdgcn_tensor_load_to_lds(g0, g1, z4, z4, 0);
#endif
}
#endif

__launch_bounds__(256)
__global__ void diffattn_fused_kernel(const float* __restrict__ x,
                                      const float* __restrict__ pool,
                                      const float* __restrict__ scl,
                                      float* __restrict__ out) {
  extern __shared__ float smem[];
  float* Ybuf0 = smem;                   // padded 16 x 1028 tile, buffer 0
  float* Ybuf1 = Ybuf0 + TILEF;          // buffer 1
  float* Sp    = Ybuf1 + TILEF;          // 8 x 16 x 16 score partials
  float* Sful  = Sp + WAVES * NT * NT;   // 16 x 16 reduced scores
  float* Pt    = Sful + NT * NT;         // 16 x 16 probabilities
  float* mrun  = Pt + NT * NT;           // 16 running max
  float* lrun  = mrun + NT;              // 16 running denom
  float* cfac  = lrun + NT;              // 16 accumulator correction

  const int tid    = threadIdx.x;
  const int wave   = tid >> 5;
  const int lane   = tid & 31;
  const int lanelo = lane & 15;
  const int mo     = (lane < 16) ? 0 : 8;  // C/D row offset per half-wave
  const int khalf  = lane >> 4;            // A/B K-half select

  const int b0 = blockIdx.x * NT;
  const int r  = blockIdx.y;
  const int a  = blockIdx.z;

  const int dw0 = wave * DPW;
  const float* sa    = scl + a * Dd;
  const float* poolr = pool + (size_t)r * Nn * Dd;

#if HAVE_TDM
  const unsigned lds_base = __builtin_amdgcn_groupstaticsize();
  const unsigned ybyte[2] = {lds_base, lds_base + (unsigned)(TILEF * 4)};
#endif

  if (tid < NT) { mrun[tid] = -INFINITY; lrun[tid] = 0.0f; }

  // Query fragments, loaded once: Xs[m][d] = x[b0+m, r, d] * scl[a, d].
  // 32-bit WMMA A(16x4): lane m = L%16; lanes 0-15 hold K={0,1},
  // lanes 16-31 hold K={2,3}.
  v2f xf[DPW / 4];
#pragma unroll
  for (int k = 0; k < DPW / 4; ++k) {
    int d = dw0 + 4 * k + 2 * khalf;
    const float* xp = x + ((size_t)(b0 + lanelo) * Rr + r) * Dd + d;
    v2f v;
    v.x = xp[0] * sa[d];
    v.y = xp[1] * sa[d + 1];
    xf[k] = v;
  }

  v8f oacc[DPW / 16];
#pragma unroll
  for (int j = 0; j < DPW / 16; ++j) oacc[j] = (v8f){0,0,0,0,0,0,0,0};

#if HAVE_TDM
  if (wave == 0) tdm_load_tile(poolr, ybyte[0]);  // preload tile 0
#endif

  for (int nt = 0; nt < NTILES; ++nt) {
    float* Yb = (nt & 1) ? Ybuf1 : Ybuf0;

#if HAVE_TDM
    if (wave == 0) {
      if (nt + 1 < NTILES) {  // async-prefetch next tile into other buffer
        tdm_load_tile(poolr + (size_t)(nt + 1) * NT * Dd, ybyte[(nt + 1) & 1]);
        __builtin_amdgcn_s_wait_tensorcnt(1);  // tile nt complete, nt+1 in flight
      } else {
        __builtin_amdgcn_s_wait_tensorcnt(0);
      }
    }
    __syncthreads();  // broadcast tile nt; prior-iteration reads already fenced
#else
    __syncthreads();
    for (int q = tid; q < NT * Dd; q += 256) {
      int n = q >> 10, d = q & (Dd - 1);
      Yb[n * YSTR + d + (d >> 8)] = poolr[(size_t)nt * NT * Dd + q];
    }
    __syncthreads();
#endif

    // ---- scores: S_w(16x16) = Xs_w(16x128) * Y_w^T(128x16), K chained ----
    v8f sacc = (v8f){0,0,0,0,0,0,0,0};
#pragma unroll
    for (int k = 0; k < DPW / 4; ++k) {
      int d = dw0 + 4 * k + 2 * khalf;
      int yd = d + (d >> 8);  // padded column index (d even -> same chunk)
      v2f bf;
      bf.x = Yb[lanelo * YSTR + yd];
      bf.y = Yb[lanelo * YSTR + yd + 1];
      sacc = __builtin_amdgcn_wmma_f32_16x16x4_f32(false, xf[k], false, bf,
                                                   (short)0, sacc, false, false);
    }
#pragma unroll
    for (int i = 0; i < 8; ++i)
      Sp[wave * NT * NT + (i + mo) * NT + lanelo] = sacc[i];
    __syncthreads();

    // ---- cross-wave K reduction (one element per thread) ----
    {
      float acc = 0.0f;
#pragma unroll
      for (int w = 0; w < WAVES; ++w) acc += Sp[w * NT * NT + tid];
      Sful[tid] = acc;  // m = tid/16, n = tid%16
    }
    __syncthreads();

    // ---- online softmax over this 16-key tile ----
    if (tid < NT) {
      int m = tid;
      float mx = Sful[m * NT];
#pragma unroll
      for (int n = 1; n < NT; ++n) mx = fmaxf(mx, Sful[m * NT + n]);
      float mold = mrun[m];
      float mnew = fmaxf(mold, mx);
      float c = __expf(mold - mnew);  // first tile: exp(-inf)=0
      float sum = 0.0f;
#pragma unroll
      for (int n = 0; n < NT; ++n) {
        float p = __expf(Sful[m * NT + n] - mnew);
        Pt[m * NT + n] = p;
        sum += p;
      }
      mrun[m] = mnew;
      lrun[m] = lrun[m] * c + sum;
      cfac[m] = c;
    }
    __syncthreads();

    // ---- rescale accumulators, then O_w += P(16x16) * Y_w(16x128) ----
    float cf[8];
#pragma unroll
    for (int i = 0; i < 8; ++i) cf[i] = cfac[i + mo];
#pragma unroll
    for (int j = 0; j < DPW / 16; ++j) {
#pragma unroll
      for (int i = 0; i < 8; ++i) oacc[j][i] *= cf[i];
#pragma unroll
      for (int kc = 0; kc < NT / 4; ++kc) {
        int kk = 4 * kc + 2 * khalf;
        v2f af;
        af.x = Pt[lanelo * NT + kk];
        af.y = Pt[lanelo * NT + kk + 1];
        int col = dw0 + 16 * j + lanelo;
        int yc = col + (col >> 8);
        v2f bf;
        bf.x = Yb[kk * YSTR + yc];
        bf.y = Yb[(kk + 1) * YSTR + yc];
        oacc[j] = __builtin_amdgcn_wmma_f32_16x16x4_f32(false, af, false, bf,
                                                        (short)0, oacc[j],
                                                        false, false);
      }
    }
    __syncthreads();  // PV reads done: next TDM issue may overwrite this buffer
  }

  // ---- normalize by softmax denominator and store (B,R,A,D) ----
  float li[8];
#pragma unroll
  for (int i = 0; i < 8; ++i) li[i] = 1.0f / lrun[i + mo];
#pragma unroll
  for (int j = 0; j < DPW / 16; ++j) {
#pragma unroll
    for (int i = 0; i < 8; ++i) {
      int m = i + mo;
      int d = dw0 + 16 * j + lanelo;
      out[(((size_t)(b0 + m) * Rr + r) * Ah + a) * Dd + d] = oacc[j][i] * li[i];
    }
  }
}

extern "C" void kernel_launch(void* const* d_in, const int* in_sizes, int n_in,
                              void* d_out, int out_size, void* d_ws, size_t ws_size,
                              hipStream_t stream) {
  (void)in_sizes; (void)n_in; (void)out_size; (void)ws_size;
  const float* top  = (const float*)d_in[0];  // (B, R, D)
  const float* pool = (const float*)d_in[1];  // (R, N, D)
  const float* wx   = (const float*)d_in[2];  // (A, D, D)
  const float* wxb  = (const float*)d_in[3];  // (D)
  const float* wy   = (const float*)d_in[4];  // (A, D, D)
  const float* wyb  = (const float*)d_in[5];  // (D)
  float* scl = (float*)d_ws;                  // (A, D) = 32 KB scratch
  float* out = (float*)d_out;                 // (B, R, A, D)

  diffattn_scale_kernel<<<(Ah * Dd + 255) / 256, 256, 0, stream>>>(wx, wxb, wy,
                                                                   wyb, scl);

  size_t smem_bytes =
      (size_t)(2 * TILEF + WAVES * NT * NT + 2 * NT * NT + 3 * NT) *
      sizeof(float);  // ~138.7 KB dynamic LDS (320 KB/WGP -> 2 WGs/WGP)
  dim3 grid(Bq / NT, Rr, Ah);  // (8, 29, 8)
  diffattn_fused_kernel<<<grid, 256, smem_bytes, stream>>>(top, pool, scl, out);
}